// InputEmbedder_50027779064397
// MI455X (gfx1250) — compile-verified
//
#include <hip/hip_runtime.h>

typedef float v2f __attribute__((ext_vector_type(2)));
typedef float v8f __attribute__((ext_vector_type(8)));
typedef float f4  __attribute__((ext_vector_type(4)));

#define N_TOK   768
#define D_TF    447
#define C_S     384
#define C_Z     128
#define TFD     831   // 447 + 384
#define REL_DIM 139   // 66 + 66 + 1 + 6

// ---------------------------------------------------------------------------
// 1) s_input = concat(target_feat, token_act) along feature dim
// ---------------------------------------------------------------------------
__global__ __launch_bounds__(256) void k_concat(const float* __restrict__ tf,
                                                const float* __restrict__ ta,
                                                float* __restrict__ s_input) {
  int idx = blockIdx.x * blockDim.x + threadIdx.x;
  const int total = N_TOK * TFD;
  if (idx >= total) return;
  int n = idx / TFD;
  int d = idx - n * TFD;
  s_input[idx] = (d < D_TF) ? tf[n * D_TF + d] : ta[n * C_S + (d - D_TF)];
}

// ---------------------------------------------------------------------------
// 2) W_posT[r, c] = W_pos[c, r]   ((128,139) -> (139,128)) for coalesced gathers
// ---------------------------------------------------------------------------
__global__ __launch_bounds__(256) void k_transpose_wpos(const float* __restrict__ wpos,
                                                        float* __restrict__ wpT) {
  int idx = blockIdx.x * blockDim.x + threadIdx.x;
  const int total = REL_DIM * C_Z;
  if (idx >= total) return;
  int r = idx / C_Z;
  int c = idx - r * C_Z;
  wpT[idx] = wpos[c * REL_DIM + r];
}

// ---------------------------------------------------------------------------
// 3) WMMA GEMM: C[m,n] = sum_d S[m,d] * W[n,d]
//    One wave32 per 16x16 output tile, V_WMMA_F32_16X16X4_F32, K-loop step 4.
//    Column tiles 0..23 -> W_single -> s_init; 24..31 -> W_left -> a_ws;
//    32..39 -> W_right -> b_ws.
// ---------------------------------------------------------------------------
__global__ __launch_bounds__(128) void k_gemm_wmma(const float* __restrict__ S,
                                                   const float* __restrict__ Wsingle,
                                                   const float* __restrict__ Wleft,
                                                   const float* __restrict__ Wright,
                                                   float* __restrict__ s_init,
                                                   float* __restrict__ a_ws,
                                                   float* __restrict__ b_ws) {
  const int NT_N = (C_S + 2 * C_Z) / 16;  // 40 column tiles
  const int NT_M = N_TOK / 16;            // 48 row tiles

  int wave = threadIdx.x >> 5;
  int lane = threadIdx.x & 31;
  int tile = blockIdx.x * 4 + wave;
  if (tile >= NT_M * NT_N) return;
  int tm = tile / NT_N;
  int tn = tile - tm * NT_N;

  const float* W;
  float* dst;
  int ldd, ncol;
  if (tn < C_S / 16) {
    W = Wsingle; dst = s_init; ldd = C_S; ncol = tn * 16;
  } else if (tn < C_S / 16 + C_Z / 16) {
    W = Wleft;   dst = a_ws;   ldd = C_Z; ncol = (tn - C_S / 16) * 16;
  } else {
    W = Wright;  dst = b_ws;   ldd = C_Z; ncol = (tn - C_S / 16 - C_Z / 16) * 16;
  }

  // f32 16x16x4 fragment layout (wave32):
  //   A: lane L holds row M=L%16, K = kbase + 2*(L/16) + {0,1}
  //   B: lane L holds col N=L%16, same K pattern
  //   C: VGPR r holds M = r + 8*(L/16), N = L%16
  int mrow  = tm * 16 + (lane & 15);
  int nrow  = ncol + (lane & 15);
  int khalf = (lane >> 4) * 2;

  const float* Sp = S + (size_t)mrow * TFD;
  const float* Wp = W + (size_t)nrow * TFD;

  __builtin_prefetch(Sp, 0, 3);  // global_prefetch_b8
  __builtin_prefetch(Wp, 0, 3);

  v8f acc = {};
  int k = 0;
  for (; k < 828; k += 4) {      // full K steps: k+3 <= 830
    int kk = k + khalf;
    v2f av, bv;
    av.x = Sp[kk]; av.y = Sp[kk + 1];
    bv.x = Wp[kk]; bv.y = Wp[kk + 1];
    acc = __builtin_amdgcn_wmma_f32_16x16x4_f32(false, av, false, bv,
                                                (short)0, acc, false, false);
  }
  {                              // tail k = 828 .. 830 (K = 831)
    int kk = k + khalf;
    v2f av, bv;
    av.x = (kk     < TFD) ? Sp[kk]     : 0.0f;
    av.y = (kk + 1 < TFD) ? Sp[kk + 1] : 0.0f;
    bv.x = (kk     < TFD) ? Wp[kk]     : 0.0f;
    bv.y = (kk + 1 < TFD) ? Wp[kk + 1] : 0.0f;
    acc = __builtin_amdgcn_wmma_f32_16x16x4_f32(false, av, false, bv,
                                                (short)0, acc, false, false);
  }

  int cbase = tm * 16 + 8 * (lane >> 4);
  int cn    = ncol + (lane & 15);
#pragma unroll
  for (int r = 0; r < 8; ++r) {
    dst[(size_t)(cbase + r) * ldd + cn] = acc[r];
  }
}

// ---------------------------------------------------------------------------
// 4) Per-(i,j) pair: z_init (128 ch, float4/lane) + rel_feat one-hots (139).
//    One wave32 per pair, 8 pairs per 256-thread block. rel_enc GEMM is
//    replaced by 4 gathered rows of W_posT (one-hot structure): 21 GFLOP -> 0.
// ---------------------------------------------------------------------------
__global__ __launch_bounds__(256) void k_pair(const float* __restrict__ a_ws,
                                              const float* __restrict__ b_ws,
                                              const float* __restrict__ wpT,   // [139,128]
                                              const float* __restrict__ wbond, // [128]
                                              const float* __restrict__ contact,
                                              const int* __restrict__ tok,
                                              const int* __restrict__ res,
                                              const int* __restrict__ asym,
                                              const int* __restrict__ ent,
                                              const int* __restrict__ sym,
                                              float* __restrict__ z_init,
                                              float* __restrict__ rel_feat) {
  int lane = threadIdx.x & 31;
  int p = blockIdx.x * 8 + (threadIdx.x >> 5);   // grid is exact: p < 768*768
  int i = p / N_TOK;
  int j = p - i * N_TOK;

  int lt = tok[i],  rt = tok[j];
  int lr = res[i],  rr = res[j];
  int la = asym[i], ra = asym[j];
  int le = ent[i],  re = ent[j];
  int ls = sym[i],  rs = sym[j];

  bool same_chain = (la == ra);
  bool same_res   = (lr == rr);
  bool same_ent   = (le == re);

  int rd = lr - rr + 32; rd = rd < 0 ? 0 : (rd > 64 ? 64 : rd);
  if (!same_chain) rd = 65;
  int td = lt - rt + 32; td = td < 0 ? 0 : (td > 64 ? 64 : td);
  if (!(same_chain && same_res)) td = 65;
  int cd = ls - rs + 2;  cd = cd < 0 ? 0 : (cd > 4 ? 4 : cd);
  if (!same_ent) cd = 5;

  float se = same_ent ? 1.0f : 0.0f;
  float ct = contact[p];

  // z_init[p, c] = a[i,c] + b[j,c] + Wpos gathers + ct * Wbond[c]
  int c0 = lane * 4;
  f4 va = *(const f4*)(a_ws + (size_t)i * C_Z + c0);
  f4 vb = *(const f4*)(b_ws + (size_t)j * C_Z + c0);
  f4 w1 = *(const f4*)(wpT + (size_t)rd * C_Z + c0);
  f4 w2 = *(const f4*)(wpT + (size_t)(66 + td) * C_Z + c0);
  f4 w3 = *(const f4*)(wpT + (size_t)132 * C_Z + c0);
  f4 w4 = *(const f4*)(wpT + (size_t)(133 + cd) * C_Z + c0);
  f4 wb = *(const f4*)(wbond + c0);
  f4 z = va + vb + w1 + w2 + w3 * se + w4 + wb * ct;
  *(f4*)(z_init + (size_t)p * C_Z + c0) = z;

  // rel_feat one-hot row, fused single write pass
  float* rf = rel_feat + (size_t)p * REL_DIM;
  for (int idx = lane; idx < REL_DIM; idx += 32) {
    float v = 0.0f;
    if (idx == rd || idx == 66 + td || idx == 133 + cd) v = 1.0f;
    else if (idx == 132) v = se;
    rf[idx] = v;
  }
}

// ---------------------------------------------------------------------------
extern "C" void kernel_launch(void* const* d_in, const int* in_sizes, int n_in,
                              void* d_out, int out_size, void* d_ws, size_t ws_size,
                              hipStream_t stream) {
  const float* target_feat = (const float*)d_in[0];
  const float* token_act   = (const float*)d_in[1];
  const float* contact     = (const float*)d_in[2];
  const int*   tok         = (const int*)d_in[3];
  const int*   res         = (const int*)d_in[4];
  const int*   asym        = (const int*)d_in[5];
  const int*   ent         = (const int*)d_in[6];
  const int*   sym         = (const int*)d_in[7];
  const float* Wsingle     = (const float*)d_in[8];
  const float* Wleft       = (const float*)d_in[9];
  const float* Wright      = (const float*)d_in[10];
  const float* Wpos        = (const float*)d_in[11];
  const float* Wbond       = (const float*)d_in[12];

  float* out      = (float*)d_out;
  float* s_input  = out;                                        // 768*831
  float* s_init   = s_input + (size_t)N_TOK * TFD;              // 768*384
  float* z_init   = s_init + (size_t)N_TOK * C_S;               // 768*768*128
  float* rel_feat = z_init + (size_t)N_TOK * N_TOK * C_Z;       // 768*768*139

  float* a_ws = (float*)d_ws;                                   // 768*128
  float* b_ws = a_ws + (size_t)N_TOK * C_Z;                     // 768*128
  float* wpT  = b_ws + (size_t)N_TOK * C_Z;                     // 139*128

  {
    int total = N_TOK * TFD;
    k_concat<<<(total + 255) / 256, 256, 0, stream>>>(target_feat, token_act, s_input);
  }
  {
    int total = REL_DIM * C_Z;
    k_transpose_wpos<<<(total + 255) / 256, 256, 0, stream>>>(Wpos, wpT);
  }
  {
    // 48*40 = 1920 tiles, 4 waves (tiles) per 128-thread block
    k_gemm_wmma<<<480, 128, 0, stream>>>(s_input, Wsingle, Wleft, Wright,
                                         s_init, a_ws, b_ws);
  }
  {
    // 768*768 pairs, 8 pairs per 256-thread block
    k_pair<<<(N_TOK * N_TOK) / 8, 256, 0, stream>>>(a_ws, b_ws, wpT, Wbond, contact,
                                                    tok, res, asym, ent, sym,
                                                    z_init, rel_feat);
  }
}